// ObjectChecklistModel_69020124447176
// MI455X (gfx1250) — compile-verified
//
#include <hip/hip_runtime.h>
#include <math.h>

typedef unsigned int u32;
typedef float v2f __attribute__((ext_vector_type(2)));
typedef float v8f __attribute__((ext_vector_type(8)));

// ---------------- problem constants ----------------
constexpr int   B    = 1024;     // batch
constexpr int   H    = 256;      // key dim
constexpr int   M    = 262144;   // memory slots
constexpr int   KNN  = 256;      // top-k
constexpr float ALPHA = 0.1f;
constexpr float TEMP  = 39.3573952f;   // max(1, log(0.2*256)/0.1)

// ---------------- algorithm constants ----------------
constexpr int NBIN  = 2048;      // sim histogram bins (top 11 ordered-float bits)
constexpr int CAP   = 2048;      // candidate cap per query
constexpr int ABIN  = 65536;     // age histogram bins (top 16 ordered-float bits)
constexpr int ACAP  = 8192;      // age candidate cap
constexpr int CHUNK = 32768;     // memory rows per GEMM block
constexpr int NCHUNK = M / CHUNK;   // 8
constexpr int QT    = 16;        // queries per GEMM block
constexpr int QSTRIDE = 260;     // padded LDS row stride (floats) to avoid bank conflicts

// ---------------- workspace layout (u32 units) ----------------
constexpr size_t OFF_HIST    = 0;
constexpr size_t OFF_AGEHIST = OFF_HIST + (size_t)B * NBIN;
constexpr size_t OFF_CCNT    = OFF_AGEHIST + ABIN;
constexpr size_t OFF_AGECNT  = OFF_CCNT + B;
constexpr size_t OFF_MSUM    = OFF_AGECNT + 1;
constexpr size_t OFF_KSUM    = OFF_MSUM + 1;
constexpr size_t OFF_AGETHR  = OFF_KSUM + 1;
constexpr size_t OFF_OWNER   = OFF_AGETHR + 1;          // per-slot last-writer tag
constexpr size_t ZEND        = OFF_OWNER + M;           // zero-init region end
constexpr size_t OFF_NQ      = ZEND;
constexpr size_t OFF_THR     = OFF_NQ + (size_t)B * H;
constexpr size_t OFF_CSIM    = OFF_THR + B;
constexpr size_t OFF_CIDX    = OFF_CSIM + (size_t)B * CAP;
constexpr size_t OFF_CORIDX  = OFF_CIDX + (size_t)B * CAP;
constexpr size_t OFF_IND     = OFF_CORIDX + B;
constexpr size_t OFF_UPDW    = OFF_IND + B;
constexpr size_t OFF_AGEV    = OFF_UPDW + B;
constexpr size_t OFF_AGEI    = OFF_AGEV + ACAP;
constexpr size_t OFF_UPD     = OFF_AGEI + ACAP;

// ---------------- helpers ----------------
__device__ __forceinline__ u32 ford(float f) {
  u32 u = __float_as_uint(f);
  return (u & 0x80000000u) ? ~u : (u | 0x80000000u);
}

__device__ __forceinline__ float blockReduceSum(float v, float* red) {
  int tid = threadIdx.x;
  red[tid] = v; __syncthreads();
  for (int s = blockDim.x >> 1; s > 0; s >>= 1) {
    if (tid < s) red[tid] += red[tid + s];
    __syncthreads();
  }
  float r = red[0]; __syncthreads();
  return r;
}

__device__ __forceinline__ void bitonic_desc(float* s, int* idx, int N, int tid, int nthr) {
  for (int k = 2; k <= N; k <<= 1) {
    for (int j = k >> 1; j > 0; j >>= 1) {
      for (int i = tid; i < N; i += nthr) {
        int ixj = i ^ j;
        if (ixj > i) {
          bool descSeg = ((i & k) == 0);
          float a = s[i], b = s[ixj];
          bool sw = descSeg ? (a < b) : (a > b);
          if (sw) {
            s[i] = b; s[ixj] = a;
            int t = idx[i]; idx[i] = idx[ixj]; idx[ixj] = t;
          }
        }
      }
      __syncthreads();
    }
  }
}

// ---------------- kernels ----------------
__global__ void k_init(u32* __restrict__ ws) {
  size_t i = (size_t)blockIdx.x * blockDim.x + threadIdx.x;
  size_t stride = (size_t)gridDim.x * blockDim.x;
  for (; i < ZEND; i += stride) ws[i] = 0u;
}

__global__ __launch_bounds__(256) void k_normalize(
    const float* __restrict__ qk, const float* __restrict__ qm,
    float* __restrict__ nq, float* __restrict__ ksum) {
  __shared__ float red[256];
  int q = blockIdx.x, tid = threadIdx.x;
  float v = qk[(size_t)q * H + tid];
  float ss = blockReduceSum(v * v, red);
  float inv = 1.0f / fmaxf(sqrtf(ss), 1e-12f);
  nq[(size_t)q * H + tid] = v * inv;
  if (tid == 0) atomicAdd(ksum, qm[q]);
}

// Fused fp32 WMMA GEMM: PASS 0 = histogram of sims, PASS 1 = gather candidates.
template <int PASS>
__global__ __launch_bounds__(256) void k_sims(
    const float* __restrict__ nq, const float* __restrict__ mk,
    u32* __restrict__ hist, const u32* __restrict__ thr,
    u32* __restrict__ ccnt, float* __restrict__ csim, u32* __restrict__ cidx) {
  extern __shared__ char smem[];
  float* qlds = (float*)smem;                                   // QT x QSTRIDE
  u32*   h    = (u32*)(smem + QT * QSTRIDE * sizeof(float));    // QT x NBIN (PASS 0)

  const int tid   = threadIdx.x;
  const int wave  = tid >> 5;
  const int lane  = tid & 31;
  const int lane16 = lane & 15;
  const int khalf  = lane >> 4;          // 0: K={0,1}, 1: K={2,3} per wmma step
  const int q0 = blockIdx.y * QT;
  const int m0 = blockIdx.x * CHUNK;

  for (int i = tid; i < QT * H; i += 256) {
    int r = i >> 8, c = i & (H - 1);
    qlds[r * QSTRIDE + c] = nq[(size_t)(q0 + r) * H + c];
  }
  if (PASS == 0) {
    for (int i = tid; i < QT * NBIN; i += 256) h[i] = 0u;
  }
  __syncthreads();

  const float* qa = qlds + lane16 * QSTRIDE + 2 * khalf;

  for (int it = 0; it < CHUNK / 128; ++it) {
    const int mrow = m0 + it * 128 + wave * 16 + lane16;   // this lane's B column / memory row
    const float* brow = mk + (size_t)mrow * H + 2 * khalf;
    __builtin_prefetch(brow + (size_t)128 * H, 0, 0);      // next 128-row tile -> global_prefetch_b8
    v8f acc = {};
#pragma unroll 8
    for (int k0 = 0; k0 < H; k0 += 4) {
      v2f a = *(const v2f*)(qa + k0);      // A: 16x4 fp32 tile slice (LDS)
      v2f b = *(const v2f*)(brow + k0);    // B: 4x16 fp32 tile slice (global)
      acc = __builtin_amdgcn_wmma_f32_16x16x4_f32(
          false, a, false, b, (short)0, acc, false, false);
    }
#pragma unroll
    for (int j = 0; j < 8; ++j) {
      float s = acc[j];
      int ql = j + 8 * khalf;              // query within tile (C layout: M=j+8*(lane/16))
      if (PASS == 0) {
        u32 bin = ford(s) >> 21;
        atomicAdd(&h[ql * NBIN + bin], 1u);
      } else {
        int q = q0 + ql;
        if (ford(s) >= thr[q]) {
          u32 pos = atomicAdd(&ccnt[q], 1u);
          if (pos < (u32)CAP) {
            csim[(size_t)q * CAP + pos] = s;
            cidx[(size_t)q * CAP + pos] = (u32)mrow;
          }
        }
      }
    }
  }

  if (PASS == 0) {
    __syncthreads();
    for (int i = tid; i < QT * NBIN; i += 256) {
      u32 c = h[i];
      if (c) atomicAdd(&hist[(size_t)(q0 + i / NBIN) * NBIN + (i & (NBIN - 1))], c);
    }
  }
}

__global__ void k_threshold(const u32* __restrict__ hist, u32* __restrict__ thr) {
  int q = blockIdx.x * blockDim.x + threadIdx.x;
  if (q >= B) return;
  const u32* hq = hist + (size_t)q * NBIN;
  u32 cum = 0; int b = NBIN - 1;
  for (; b >= 0; --b) { cum += hq[b]; if (cum >= (u32)KNN) break; }
  thr[q] = (b >= 0) ? ((u32)b << 21) : 0u;
}

__global__ __launch_bounds__(256) void k_topk(
    const float* __restrict__ nq, const float* __restrict__ mk,
    const float* __restrict__ mv, const float* __restrict__ qv,
    const float* __restrict__ qm, const int* __restrict__ recent,
    const u32* __restrict__ ccnt, const float* __restrict__ csim,
    const u32* __restrict__ cidx,
    float* __restrict__ out_rv, int* __restrict__ correct_idx,
    float* __restrict__ indicator, float* __restrict__ msum) {
  __shared__ float ssim[CAP];
  __shared__ int   sidx[CAP];
  __shared__ float red[256];
  __shared__ int s_fc, s_fw, s_gc, s_gw;
  __shared__ float s_ind;
  const int q = blockIdx.x, tid = threadIdx.x;
  u32 cn = ccnt[q];
  int n = (cn < (u32)CAP) ? (int)cn : CAP;
  for (int i = tid; i < CAP; i += 256) {
    if (i < n) { ssim[i] = csim[(size_t)q * CAP + i]; sidx[i] = (int)cidx[(size_t)q * CAP + i]; }
    else       { ssim[i] = -INFINITY; sidx[i] = -1; }
  }
  __syncthreads();
  bitonic_desc(ssim, sidx, CAP, tid, 256);

  // softmax read-out over top-256 (tid == knn index)
  float s0 = ssim[0];
  float w  = expf((ssim[tid] - s0) * TEMP);
  int   id = sidx[tid];
  float val = (id >= 0) ? mv[id] : 0.0f;
  float S  = blockReduceSum(w, red);
  float WV = blockReduceSum(w * val, red);
  if (tid == 0) out_rv[q] = WV / S;
  __syncthreads();

  if (tid == 0) { sidx[256] = recent[0]; sidx[257] = recent[1]; s_fc = 258; s_fw = 258; }
  __syncthreads();
  float qvq = qv[q];
  for (int i = tid; i < 258; i += 256) {
    int g = sidx[i]; if (g < 0) g = 0;
    bool c = (mv[g] == qvq);
    if (c) atomicMin(&s_fc, i); else atomicMin(&s_fw, i);
    if (i == 0) s_ind = c ? 1.0f : 0.0f;
  }
  __syncthreads();
  if (tid == 0) {
    int fc = (s_fc < 258) ? s_fc : 0;
    int fw = (s_fw < 258) ? s_fw : 0;
    int gc = sidx[fc]; if (gc < 0) gc = 0;
    int gw = sidx[fw]; if (gw < 0) gw = 0;
    s_gc = gc; s_gw = gw;
    correct_idx[q] = gc;
    indicator[q]   = s_ind;
  }
  __syncthreads();
  float nv = nq[(size_t)q * H + tid];
  float dc = blockReduceSum(nv * mk[(size_t)s_gc * H + tid], red);
  float dw = blockReduceSum(nv * mk[(size_t)s_gw * H + tid], red);
  if (tid == 0) {
    float margin = fmaxf(0.0f, dw - dc + ALPHA);
    atomicAdd(msum, margin * qm[q]);
  }
}

__global__ void k_age_hist(const float* __restrict__ ma, const float* __restrict__ an,
                           u32* __restrict__ ah) {
  int i = blockIdx.x * blockDim.x + threadIdx.x;
  if (i >= M) return;
  atomicAdd(&ah[ford(ma[i] + an[i]) >> 16], 1u);
}

// Parallel segmented scan for the age threshold (256 segments of 256 bins).
__global__ __launch_bounds__(256) void k_age_thresh(const u32* __restrict__ ah,
                                                    u32* __restrict__ athr) {
  __shared__ u32 seg[256];
  constexpr int SEG = ABIN / 256;   // 256 bins per segment
  int tid = threadIdx.x;
  u32 sum = 0;
  for (int b = 0; b < SEG; ++b) sum += ah[tid * SEG + b];
  seg[tid] = sum;
  __syncthreads();
  if (tid == 0) {
    u32 cum = 0; int s = 255;
    for (; s >= 0; --s) { cum += seg[s]; if (cum >= (u32)B) break; }
    if (s < 0) { athr[0] = 0u; return; }
    u32 cumAbove = 0;
    for (int t = s + 1; t < 256; ++t) cumAbove += seg[t];
    u32 c2 = cumAbove; int b = (s + 1) * SEG - 1;
    for (; b >= s * SEG; --b) { c2 += ah[b]; if (c2 >= (u32)B) break; }
    athr[0] = ((u32)b << 16);
  }
}

__global__ void k_age_gather(const float* __restrict__ ma, const float* __restrict__ an,
                             const u32* __restrict__ athr, u32* __restrict__ acnt,
                             float* __restrict__ av, int* __restrict__ ai) {
  int i = blockIdx.x * blockDim.x + threadIdx.x;
  if (i >= M) return;
  float a = ma[i] + an[i];
  if (ford(a) >= athr[0]) {
    u32 p = atomicAdd(acnt, 1u);
    if (p < (u32)ACAP) { av[p] = a; ai[p] = i; }
  }
}

__global__ __launch_bounds__(256) void k_age_sort(const u32* __restrict__ acnt,
                                                  const float* __restrict__ av,
                                                  const int* __restrict__ ai,
                                                  int* __restrict__ updw) {
  extern __shared__ char smem[];
  float* sv = (float*)smem;
  int*   si = (int*)(smem + ACAP * sizeof(float));
  int tid = threadIdx.x;
  u32 cn = acnt[0];
  int n = (cn < (u32)ACAP) ? (int)cn : ACAP;
  for (int i = tid; i < ACAP; i += 256) {
    if (i < n) { sv[i] = av[i]; si[i] = ai[i]; }
    else       { sv[i] = -INFINITY; si[i] = 0; }
  }
  __syncthreads();
  bitonic_desc(sv, si, ACAP, tid, 256);
  for (int i = tid; i < B; i += 256) updw[i] = si[i];
}

// 256 MB table copy via the CDNA5 async Global<->LDS path.
// Loads: b128 (src is 16B aligned). Stores: b32 (dst is only 4B aligned).
__global__ __launch_bounds__(256) void k_copy_keys_async(const float* __restrict__ mk,
                                                         float* __restrict__ ok) {
  __shared__ __align__(16) float buf[256 * 16];   // 64 bytes per thread
  const size_t N = (size_t)M * H;
  unsigned lds = (unsigned)(size_t)(&buf[threadIdx.x * 16]);
  size_t i = ((size_t)blockIdx.x * blockDim.x + threadIdx.x) * 16;
  const size_t stride = (size_t)gridDim.x * blockDim.x * 16;
  for (; i < N; i += stride) {
    unsigned long long gsrc = (unsigned long long)(const void*)(mk + i);
    unsigned long long gdst = (unsigned long long)(void*)(ok + i);
    asm volatile(
        "global_load_async_to_lds_b128 %0, %1, off\n\t"
        "global_load_async_to_lds_b128 %0, %1, off offset:16\n\t"
        "global_load_async_to_lds_b128 %0, %1, off offset:32\n\t"
        "global_load_async_to_lds_b128 %0, %1, off offset:48\n\t"
        "s_wait_asynccnt 0x0"
        :: "v"(lds), "v"(gsrc) : "memory");
    asm volatile(
        "global_store_async_from_lds_b32 %0, %1, off\n\t"
        "global_store_async_from_lds_b32 %0, %1, off offset:4\n\t"
        "global_store_async_from_lds_b32 %0, %1, off offset:8\n\t"
        "global_store_async_from_lds_b32 %0, %1, off offset:12\n\t"
        "global_store_async_from_lds_b32 %0, %1, off offset:16\n\t"
        "global_store_async_from_lds_b32 %0, %1, off offset:20\n\t"
        "global_store_async_from_lds_b32 %0, %1, off offset:24\n\t"
        "global_store_async_from_lds_b32 %0, %1, off offset:28\n\t"
        "global_store_async_from_lds_b32 %0, %1, off offset:32\n\t"
        "global_store_async_from_lds_b32 %0, %1, off offset:36\n\t"
        "global_store_async_from_lds_b32 %0, %1, off offset:40\n\t"
        "global_store_async_from_lds_b32 %0, %1, off offset:44\n\t"
        "global_store_async_from_lds_b32 %0, %1, off offset:48\n\t"
        "global_store_async_from_lds_b32 %0, %1, off offset:52\n\t"
        "global_store_async_from_lds_b32 %0, %1, off offset:56\n\t"
        "global_store_async_from_lds_b32 %0, %1, off offset:60\n\t"
        "s_wait_asynccnt 0x0"
        :: "v"(gdst), "v"(lds) : "memory");
  }
}

__global__ void k_copy_small(const float* __restrict__ mv, const float* __restrict__ ma,
                             float* __restrict__ ov, float* __restrict__ oa) {
  int i = blockIdx.x * blockDim.x + threadIdx.x;
  if (i < M) { ov[i] = mv[i]; oa[i] = ma[i] + 1.0f; }
}

// Deterministic parallel scatter: highest query index wins each slot.
__global__ void k_prepare_upd(const float* __restrict__ qm, const int* __restrict__ coridx,
                              const float* __restrict__ indic, const int* __restrict__ updw,
                              int* __restrict__ updarr, u32* __restrict__ owner) {
  int q = blockIdx.x * blockDim.x + threadIdx.x;
  if (q >= B) return;
  int u = (indic[q] > 0.0f) ? coridx[q] : updw[q];
  updarr[q] = u;
  if (qm[q] > 0.0f) atomicMax(&owner[u], (u32)(q + 1));
}

__global__ __launch_bounds__(256) void k_scatter_rows(
    const float* __restrict__ nq, const float* __restrict__ mk,
    const float* __restrict__ qv, const float* __restrict__ qm,
    const int* __restrict__ coridx, const float* __restrict__ indic,
    const int* __restrict__ updarr, const u32* __restrict__ owner,
    float* __restrict__ ok, float* __restrict__ ov, float* __restrict__ oa) {
  __shared__ float red[256];
  int q = blockIdx.x, tid = threadIdx.x;
  if (qm[q] <= 0.0f) return;                 // uniform per block
  int u = updarr[q];
  if (owner[u] != (u32)(q + 1)) return;      // a later query owns this slot
  float nv = nq[(size_t)q * H + tid];
  float x;
  if (indic[q] > 0.0f) {
    x = mk[(size_t)coridx[q] * H + tid] + nv;
    float ss = blockReduceSum(x * x, red);
    x *= 1.0f / fmaxf(sqrtf(ss), 1e-12f);
  } else {
    x = nv;
  }
  ok[(size_t)u * H + tid] = x;
  if (tid == 0) { ov[u] = qv[q]; oa[u] = 0.0f; }
}

__global__ void k_finalize(const float* __restrict__ qv, const float* __restrict__ qm,
                           const int* __restrict__ updarr,
                           const float* __restrict__ msum, const float* __restrict__ ksum,
                           float* __restrict__ orec, float* __restrict__ oloss) {
  if (threadIdx.x != 0 || blockIdx.x != 0) return;
  float bestp = -INFINITY; int bp = 0;
  float bestn =  INFINITY; int bn = 0;
  for (int q = 0; q < B; ++q) {
    float keep = qm[q];
    float vp = (keep > 0.0f) ? qv[q] : -INFINITY;
    float vn = (keep > 0.0f) ? qv[q] :  INFINITY;
    if (vp > bestp) { bestp = vp; bp = q; }
    if (vn < bestn) { bestn = vn; bn = q; }
  }
  orec[0] = (float)updarr[bp];
  orec[1] = (float)updarr[bn];
  oloss[0] = msum[0] / ksum[0];
}

// ---------------- launcher ----------------
extern "C" void kernel_launch(void* const* d_in, const int* in_sizes, int n_in,
                              void* d_out, int out_size, void* d_ws, size_t ws_size,
                              hipStream_t stream) {
  const float* qk     = (const float*)d_in[0];
  const float* qv     = (const float*)d_in[1];
  const float* qm     = (const float*)d_in[2];
  const float* mk     = (const float*)d_in[3];
  const float* mv     = (const float*)d_in[4];
  const float* ma     = (const float*)d_in[5];
  const int*   recent = (const int*)  d_in[6];
  const float* an     = (const float*)d_in[7];

  float* out  = (float*)d_out;
  float* o_rv   = out;                       // [B]
  float* o_loss = out + B;                   // [1]
  float* o_keys = out + B + 1;               // [M*H]
  float* o_vals = o_keys + (size_t)M * H;    // [M]
  float* o_ages = o_vals + M;                // [M]
  float* o_rec  = o_ages + M;                // [2]

  u32* ws = (u32*)d_ws;
  u32*   hist   = ws + OFF_HIST;
  u32*   ahist  = ws + OFF_AGEHIST;
  u32*   ccnt   = ws + OFF_CCNT;
  u32*   acnt   = ws + OFF_AGECNT;
  float* msum   = (float*)(ws + OFF_MSUM);
  float* ksum   = (float*)(ws + OFF_KSUM);
  u32*   athr   = ws + OFF_AGETHR;
  u32*   owner  = ws + OFF_OWNER;
  float* nqp    = (float*)(ws + OFF_NQ);
  u32*   thr    = ws + OFF_THR;
  float* csim   = (float*)(ws + OFF_CSIM);
  u32*   cidx   = ws + OFF_CIDX;
  int*   coridx = (int*)(ws + OFF_CORIDX);
  float* indic  = (float*)(ws + OFF_IND);
  int*   updw   = (int*)(ws + OFF_UPDW);
  float* agev   = (float*)(ws + OFF_AGEV);
  int*   agei   = (int*)(ws + OFF_AGEI);
  int*   updarr = (int*)(ws + OFF_UPD);

  k_init<<<2048, 256, 0, stream>>>(ws);
  k_normalize<<<B, 256, 0, stream>>>(qk, qm, nqp, ksum);

  size_t smem0 = (size_t)QT * QSTRIDE * sizeof(float) + (size_t)QT * NBIN * sizeof(u32);
  size_t smem1 = (size_t)QT * QSTRIDE * sizeof(float);
  k_sims<0><<<dim3(NCHUNK, B / QT), 256, smem0, stream>>>(nqp, mk, hist, thr, ccnt, csim, cidx);
  k_threshold<<<B / 256, 256, 0, stream>>>(hist, thr);
  k_sims<1><<<dim3(NCHUNK, B / QT), 256, smem1, stream>>>(nqp, mk, hist, thr, ccnt, csim, cidx);

  k_topk<<<B, 256, 0, stream>>>(nqp, mk, mv, qv, qm, recent, ccnt, csim, cidx,
                                o_rv, coridx, indic, msum);

  k_age_hist<<<M / 256, 256, 0, stream>>>(ma, an, ahist);
  k_age_thresh<<<1, 256, 0, stream>>>(ahist, athr);
  k_age_gather<<<M / 256, 256, 0, stream>>>(ma, an, athr, acnt, agev, agei);
  k_age_sort<<<1, 256, (size_t)ACAP * 8, stream>>>(acnt, agev, agei, updw);

  k_copy_keys_async<<<16384, 256, 0, stream>>>(mk, o_keys);
  k_copy_small<<<M / 256, 256, 0, stream>>>(mv, ma, o_vals, o_ages);

  k_prepare_upd<<<B / 256, 256, 0, stream>>>(qm, coridx, indic, updw, updarr, owner);
  k_scatter_rows<<<B, 256, 0, stream>>>(nqp, mk, qv, qm, coridx, indic, updarr, owner,
                                        o_keys, o_vals, o_ages);
  k_finalize<<<1, 32, 0, stream>>>(qv, qm, updarr, msum, ksum, o_rec, o_loss);
}